// GCN_encoder_32023276159006
// MI455X (gfx1250) — compile-verified
//
#include <hip/hip_runtime.h>

#define N_NODES 50000
#define N_EDGES 1000000

typedef __attribute__((ext_vector_type(2))) float v2f;
typedef __attribute__((ext_vector_type(8))) float v8f;

// ---------------- elementwise helpers ----------------

__global__ void zero_f32(float* __restrict__ p, int n) {
  int i = blockIdx.x * blockDim.x + threadIdx.x;
  if (i < n) p[i] = 0.0f;
}

__global__ void edge_degrees(const int* __restrict__ src, const int* __restrict__ dst,
                             float* __restrict__ degO, float* __restrict__ degI, int E) {
  int e = blockIdx.x * blockDim.x + threadIdx.x;
  if (e < E) {
    atomicAdd(&degO[src[e]], 1.0f);
    atomicAdd(&degI[dst[e]], 1.0f);
  }
}

__global__ void deg_to_norm(float* __restrict__ a, int n) {
  int i = blockIdx.x * blockDim.x + threadIdx.x;
  if (i < n) a[i] = rsqrtf(fmaxf(a[i], 1.0f));
}

// y[r, :] = x[r, :] * s[r]
__global__ void scale_rows(const float* __restrict__ x, const float* __restrict__ s,
                           float* __restrict__ y, int n, int d) {
  int i = blockIdx.x * blockDim.x + threadIdx.x;
  if (i < n * d) y[i] = x[i] * s[i / d];
}

// segment-sum over edges: agg[dst[e], :] += xs[src[e], :]
// one thread per (edge, 4-feature chunk); 128-bit load + 4 global f32 atomics
__global__ void edge_scatter(const float* __restrict__ xs, const int* __restrict__ src,
                             const int* __restrict__ dst, float* __restrict__ agg,
                             int E, int d) {
  int dq = d >> 2;
  int t = blockIdx.x * blockDim.x + threadIdx.x;
  if (t >= E * dq) return;
  int e = t / dq;
  int c = t - e * dq;
  const float4* vp = reinterpret_cast<const float4*>(xs + (size_t)src[e] * d);
  float4 v = vp[c];
  float* o = agg + (size_t)dst[e] * d + (c << 2);
  atomicAdd(o + 0, v.x);
  atomicAdd(o + 1, v.y);
  atomicAdd(o + 2, v.z);
  atomicAdd(o + 3, v.w);
}

// ---------------- fused (agg * norm_dst) @ W + b via V_WMMA_F32_16X16X4_F32 ----------------
// One wave per 16x16 output tile. grid = (N/16, d_out/16), block = 32 (full wave32, EXEC all 1s).
// mode bit0: relu(out0); bit1: out0 *= norm_src[row]; bit2: also write out1 = relu(v)*norm_src.
__global__ void gemm_wmma_f32(const float* __restrict__ agg,
                              const float* __restrict__ norm_dst,
                              const float* __restrict__ norm_src,
                              const float* __restrict__ W,     // [d_in, d_out] row-major
                              const float* __restrict__ bias,  // [d_out]
                              float* __restrict__ out0,
                              float* __restrict__ out1,
                              int d_in, int d_out, int mode) {
  const int lane   = threadIdx.x;      // 0..31
  const int mlocal = lane & 15;
  const int khalf  = lane >> 4;        // 0: K={k,k+1}; 1: K={k+2,k+3}
  const int rowBlk = blockIdx.x;
  const int colBlk = blockIdx.y;

  const int arow_i = rowBlk * 16 + mlocal;          // A-matrix row for this lane
  const int col    = colBlk * 16 + mlocal;          // B/D column for this lane
  const float nd   = norm_dst[arow_i];
  const float* arow = agg + (size_t)arow_i * d_in;

  v8f c = {};
  for (int k0 = 0; k0 < d_in; k0 += 4) {
    const int ka = k0 + (khalf << 1);
    v2f a, b;
    a.x = arow[ka] * nd;                               // fold norm_dst into A
    a.y = arow[ka + 1] * nd;
    b.x = W[(size_t)ka * d_out + col];
    b.y = W[(size_t)(ka + 1) * d_out + col];
    c = __builtin_amdgcn_wmma_f32_16x16x4_f32(
        /*neg_a=*/false, a, /*neg_b=*/false, b,
        /*c_mod=*/(short)0, c, /*reuse_a=*/false, /*reuse_b=*/false);
  }

  const float bval = bias[col];
#pragma unroll
  for (int v = 0; v < 8; ++v) {
    const int r = rowBlk * 16 + v + (khalf << 3);     // D row per ISA C/D layout
    float val = c[v] + bval;
    float m = val;
    if (mode & 1) m = fmaxf(m, 0.0f);
    if (mode & 2) m *= norm_src[r];
    out0[(size_t)r * d_out + col] = m;
    if (mode & 4) {
      out1[(size_t)r * d_out + col] = fmaxf(val, 0.0f) * norm_src[r];
    }
  }
}

// ---------------- driver ----------------

static inline unsigned blks(long long n, int tb) {
  return (unsigned)((n + tb - 1) / tb);
}

extern "C" void kernel_launch(void* const* d_in, const int* in_sizes, int n_in,
                              void* d_out, int out_size, void* d_ws, size_t ws_size,
                              hipStream_t stream) {
  const float* in_feat = (const float*)d_in[0];
  const float* W1 = (const float*)d_in[1];
  const float* b1 = (const float*)d_in[2];
  const float* W2 = (const float*)d_in[3];
  const float* b2 = (const float*)d_in[4];
  const float* W3 = (const float*)d_in[5];
  const float* b3 = (const float*)d_in[6];
  const int* src = (const int*)d_in[7];
  const int* dst = (const int*)d_in[8];

  float* out = (float*)d_out;
  float* embed_out = out;                               // [N, 32]
  float* final_out = out + (size_t)N_NODES * 32;        // [N, 16]

  float* ws = (float*)d_ws;
  float* norm_src = ws;                                 // [N] (deg_out -> rsqrt)
  float* norm_dst = ws + N_NODES;                       // [N] (deg_in  -> rsqrt)
  float* bufX = ws + 2 * (size_t)N_NODES;               // [N,64] gather input
  float* bufY = bufX + 64 * (size_t)N_NODES;            // [N,64] gather input
  float* bufAgg = bufY + 64 * (size_t)N_NODES;          // [N,64] segment-sum accumulator

  const int TB = 256;

  // degrees -> norms (both buffers contiguous: zero + rsqrt in one sweep)
  zero_f32<<<blks(2LL * N_NODES, TB), TB, 0, stream>>>(norm_src, 2 * N_NODES);
  edge_degrees<<<blks(N_EDGES, TB), TB, 0, stream>>>(src, dst, norm_src, norm_dst, N_EDGES);
  deg_to_norm<<<blks(2LL * N_NODES, TB), TB, 0, stream>>>(norm_src, 2 * N_NODES);

  // ---- layer 1: h1s = relu((A(in_feat*ns))*ndst @ W1 + b1) * ns  (next gather input) ----
  scale_rows<<<blks(64LL * N_NODES, TB), TB, 0, stream>>>(in_feat, norm_src, bufX, N_NODES, 64);
  zero_f32<<<blks(64LL * N_NODES, TB), TB, 0, stream>>>(bufAgg, N_NODES * 64);
  edge_scatter<<<blks(16LL * N_EDGES, TB), TB, 0, stream>>>(bufX, src, dst, bufAgg, N_EDGES, 64);
  {
    dim3 g(N_NODES / 16, 64 / 16);
    gemm_wmma_f32<<<g, 32, 0, stream>>>(bufAgg, norm_dst, norm_src, W1, b1,
                                        bufY, nullptr, 64, 64, /*relu|scale*/ 3);
  }

  // ---- layer 2: embed = agg*ndst @ W2 + b2 (to d_out); also relu(embed)*ns for layer 3 ----
  zero_f32<<<blks(64LL * N_NODES, TB), TB, 0, stream>>>(bufAgg, N_NODES * 64);
  edge_scatter<<<blks(16LL * N_EDGES, TB), TB, 0, stream>>>(bufY, src, dst, bufAgg, N_EDGES, 64);
  {
    dim3 g(N_NODES / 16, 32 / 16);
    gemm_wmma_f32<<<g, 32, 0, stream>>>(bufAgg, norm_dst, norm_src, W2, b2,
                                        embed_out, bufX, 64, 32, /*dual out*/ 4);
  }

  // ---- layer 3: out = agg*ndst @ W3 + b3 ----
  zero_f32<<<blks(32LL * N_NODES, TB), TB, 0, stream>>>(bufAgg, N_NODES * 32);
  edge_scatter<<<blks(8LL * N_EDGES, TB), TB, 0, stream>>>(bufX, src, dst, bufAgg, N_EDGES, 32);
  {
    dim3 g(N_NODES / 16, 16 / 16);
    gemm_wmma_f32<<<g, 32, 0, stream>>>(bufAgg, norm_dst, norm_src, W3, b3,
                                        final_out, nullptr, 32, 16, 0);
  }
}